// VN_Transformer_62388694942409
// MI455X (gfx1250) — compile-verified
//
#include <hip/hip_runtime.h>
#include <cstdint>
#include <cstddef>

// Problem constants (from reference)
#define Bn   8
#define Cn   64
#define Nn   8192
#define Hn   4
#define Dn   64
#define Rn   192          // 3 * C  (combined (axis, channel) index)
#define CH1  16           // f-chunks for gram kernel  -> 512 f per WG
#define SUB1 8            // sub-chunks per WG (512/64)
#define FS   64           // f per LDS sub-chunk
#define CH3  16           // f-chunks for apply kernel -> 512 f per WG
#define SUB3 8

typedef __attribute__((ext_vector_type(2))) float        v2f;
typedef __attribute__((ext_vector_type(8))) float        v8f;
typedef __attribute__((ext_vector_type(4))) unsigned int u32x4;
typedef __attribute__((ext_vector_type(8))) int          i32x8;
typedef __attribute__((ext_vector_type(4))) int          i32x4;

// ---------------------------------------------------------------------------
// Tensor Data Mover: 2D tile load (tile_d1 lines of tile_d0 contiguous fp32,
// line stride d0_stride elements) from global into LDS at lds_byte.
// D# packing per CDNA5 ISA.
// ---------------------------------------------------------------------------
static __device__ __forceinline__ void tdm_load_2d(unsigned int lds_byte,
                                                   const float* gptr,
                                                   unsigned int tile_d0,
                                                   unsigned int tile_d1,
                                                   unsigned long long d0_stride)
{
    unsigned long long ga = (unsigned long long)(uintptr_t)gptr;
    u32x4 g0;
    g0[0] = 1u;                                        // count=1, user mode
    g0[1] = lds_byte;                                  // lds_addr
    g0[2] = (unsigned int)(ga & 0xFFFFFFFFull);        // global_addr lo
    g0[3] = (unsigned int)((ga >> 32) & 0x1FFFFFFull)  // global_addr hi
          | (2u << 30);                                // type = 2 ("image")
    const unsigned int td0 = 1u << 20;                 // generous tensor dims
    const unsigned int td1 = 1u << 20;                 // (tiles never go OOB)
    i32x8 g1;
    g1[0] = (int)(2u << 16);                                        // data_size=4B
    g1[1] = (int)((td0 & 0xFFFFu) << 16);                           // tensor_dim0 lo16
    g1[2] = (int)(((td0 >> 16) & 0xFFFFu) | ((td1 & 0xFFFFu) << 16));
    g1[3] = (int)(((td1 >> 16) & 0xFFFFu) | ((tile_d0 & 0xFFFFu) << 16)); // tile_dim0
    g1[4] = (int)(tile_d1 & 0xFFFFu);                               // tile_dim1, tile_dim2=0
    g1[5] = (int)(unsigned int)(d0_stride & 0xFFFFFFFFull);         // dim0_stride lo32
    g1[6] = (int)(unsigned int)((d0_stride >> 32) & 0xFFFFull);     // dim0_stride hi16
    g1[7] = 0;
    i32x4 gz = {0, 0, 0, 0};
#if __has_include(<hip/amd_detail/amd_gfx1250_TDM.h>)
    i32x8 gz8 = {0, 0, 0, 0, 0, 0, 0, 0};
    __builtin_amdgcn_tensor_load_to_lds(g0, g1, gz, gz, gz8, 0);
#else
    __builtin_amdgcn_tensor_load_to_lds(g0, g1, gz, gz, 0);
#endif
}

// ---------------------------------------------------------------------------
// Kernel 1: per-batch Gram partials  S_b += F_b F_b^T  over a 512-token chunk.
// F_b[r=(n*64+c), f] = feat[b,c,f,n].  12x12 tiles of v_wmma_f32_16x16x4_f32.
// ---------------------------------------------------------------------------
__global__ __launch_bounds__(256) void k_gram(const float* __restrict__ feat,
                                              float* __restrict__ Spart)
{
    __shared__ float As[2][Cn][3 * FS];   // [buf][c][f_local*3 + axis]
    const int b = blockIdx.y, chunk = blockIdx.x, tid = threadIdx.x;
    const int wv = tid >> 5, lane = tid & 31, half = lane >> 4, l16 = lane & 15;
    const int waveR = wv >> 1, waveC = wv & 1;          // 4 x 2 wave grid
    const int fchunk = Nn / CH1;                        // 512
    const unsigned long long rowstride = (unsigned long long)Nn * 3;

    v8f acc[3][6];
    #pragma unroll
    for (int i = 0; i < 3; ++i)
        #pragma unroll
        for (int j = 0; j < 6; ++j)
            acc[i][j] = (v8f){0.f, 0.f, 0.f, 0.f, 0.f, 0.f, 0.f, 0.f};

    if (wv == 0) {      // prologue: TDM load of sub-chunk 0
        const float* g = feat + (((size_t)b * Cn) * Nn + (size_t)(chunk * fchunk)) * 3;
        tdm_load_2d((unsigned int)(uintptr_t)&As[0][0][0], g, 3 * FS, Cn, rowstride);
    }
    for (int s = 0; s < SUB1; ++s) {
        if (wv == 0) {
            if (s + 1 < SUB1) {
                const float* g = feat +
                    (((size_t)b * Cn) * Nn + (size_t)(chunk * fchunk + (s + 1) * FS)) * 3;
                tdm_load_2d((unsigned int)(uintptr_t)&As[(s + 1) & 1][0][0],
                            g, 3 * FS, Cn, rowstride);
                __builtin_amdgcn_s_wait_tensorcnt(1);
            } else {
                __builtin_amdgcn_s_wait_tensorcnt(0);
            }
        }
        __syncthreads();
        const float (*F)[3 * FS] = As[s & 1];
        #pragma unroll 4
        for (int k0 = 0; k0 < FS; k0 += 4) {
            const int kk = k0 + 2 * half;               // K per A/B VGPR pattern
            v2f a[3], bf[6];
            #pragma unroll
            for (int i = 0; i < 3; ++i) {
                const int r = (waveR * 3 + i) * 16 + l16;
                const int c = r & 63, n = r >> 6;
                a[i].x = F[c][kk * 3 + n];
                a[i].y = F[c][(kk + 1) * 3 + n];
            }
            #pragma unroll
            for (int j = 0; j < 6; ++j) {
                const int r2 = (waveC * 6 + j) * 16 + l16;
                const int c2 = r2 & 63, n2 = r2 >> 6;
                bf[j].x = F[c2][kk * 3 + n2];
                bf[j].y = F[c2][(kk + 1) * 3 + n2];
            }
            #pragma unroll
            for (int i = 0; i < 3; ++i)
                #pragma unroll
                for (int j = 0; j < 6; ++j)
                    acc[i][j] = __builtin_amdgcn_wmma_f32_16x16x4_f32(
                        false, a[i], false, bf[j], (short)0, acc[i][j], false, false);
        }
        __syncthreads();
    }
    float* dst = Spart + ((size_t)(b * CH1 + chunk)) * Rn * Rn;
    #pragma unroll
    for (int i = 0; i < 3; ++i)
        #pragma unroll
        for (int j = 0; j < 6; ++j) {
            const int rbase = (waveR * 3 + i) * 16 + 8 * half;
            const int col   = (waveC * 6 + j) * 16 + l16;
            #pragma unroll
            for (int v = 0; v < 8; ++v)
                dst[(size_t)(rbase + v) * Rn + col] = acc[i][j][v];
        }
}

// ---------------------------------------------------------------------------
// Kernel 2 (tiny, 1 WG per batch): reduce Gram partials, build M_h = Wq_h^T Wk_h,
// attn = <M_h, S(n,m)> -> softmax(/sqrt(192)), P_h = Wout_h Wv_h,
// U_b[(n,co),(m,ci)] = sum_h attn*P_h + I (residual folded in).
// ---------------------------------------------------------------------------
__global__ __launch_bounds__(256) void k_attn_u(const float* __restrict__ Spart,
                                                const float* __restrict__ Wq,
                                                const float* __restrict__ Wk,
                                                const float* __restrict__ Wv,
                                                const float* __restrict__ Wout,
                                                float* __restrict__ U)
{
    __shared__ float Sld[Rn * Rn];     // 147456 B: Gram
    __shared__ float Ml[Hn * Cn * Cn]; //  65536 B: M_h, later reused for P_h
    __shared__ float attnS[Hn * 9];
    const int b = blockIdx.x, tid = threadIdx.x;

    for (int idx = tid; idx < Rn * Rn; idx += 256) {   // reduce 16 partials
        float s = 0.f;
        for (int ch = 0; ch < CH1; ++ch)
            s += Spart[((size_t)(b * CH1 + ch)) * Rn * Rn + idx];
        Sld[idx] = s;
    }
    for (int idx = tid; idx < Hn * Cn * Cn; idx += 256) {   // M_h[c,c2]
        const int h = idx >> 12, c = (idx >> 6) & 63, c2 = idx & 63;
        float s = 0.f;
        for (int d = 0; d < Dn; ++d)
            s += Wq[(h * Dn + d) * Cn + c] * Wk[(h * Dn + d) * Cn + c2];
        Ml[idx] = s;
    }
    __syncthreads();
    if (tid < Hn * 9) {                 // attn[h][n][m] = <M_h, S block(n,m)>
        const int h = tid / 9, n = (tid % 9) / 3, m = tid % 3;
        float s = 0.f;
        for (int c = 0; c < Cn; ++c)
            for (int c2 = 0; c2 < Cn; ++c2)
                s += Ml[(h << 12) + (c << 6) + c2] *
                     Sld[(n * 64 + c) * Rn + (m * 64 + c2)];
        attnS[tid] = s;
    }
    __syncthreads();
    if (tid < Hn * 3) {                 // softmax over m, scale 1/sqrt(D*3)
        const int h = tid / 3, n = tid % 3;
        const float scale = 0.07216878364870322f;   // 1/sqrt(192)
        float e0 = attnS[h * 9 + n * 3 + 0] * scale;
        float e1 = attnS[h * 9 + n * 3 + 1] * scale;
        float e2 = attnS[h * 9 + n * 3 + 2] * scale;
        const float mx = fmaxf(e0, fmaxf(e1, e2));
        e0 = expf(e0 - mx); e1 = expf(e1 - mx); e2 = expf(e2 - mx);
        const float inv = 1.f / (e0 + e1 + e2);
        attnS[h * 9 + n * 3 + 0] = e0 * inv;
        attnS[h * 9 + n * 3 + 1] = e1 * inv;
        attnS[h * 9 + n * 3 + 2] = e2 * inv;
    }
    __syncthreads();
    for (int idx = tid; idx < Hn * Cn * Cn; idx += 256) {   // P_h[co,ci] (reuse Ml)
        const int h = idx >> 12, co = (idx >> 6) & 63, ci = idx & 63;
        float s = 0.f;
        for (int d = 0; d < Dn; ++d)
            s += Wout[co * (Hn * Dn) + h * Dn + d] * Wv[(h * Dn + d) * Cn + ci];
        Ml[idx] = s;
    }
    __syncthreads();
    for (int idx = tid; idx < Rn * Rn; idx += 256) {        // U (with identity)
        const int r = idx / Rn, r2 = idx - r * Rn;
        const int n = r >> 6, co = r & 63, m = r2 >> 6, ci = r2 & 63;
        float u = (r == r2) ? 1.f : 0.f;
        #pragma unroll
        for (int h = 0; h < Hn; ++h)
            u += attnS[h * 9 + n * 3 + m] * Ml[(h << 12) + (co << 6) + ci];
        U[(size_t)b * Rn * Rn + idx] = u;
    }
}

// ---------------------------------------------------------------------------
// Kernel 3: out = U_b @ F_b.  U resident in LDS, feat streamed via TDM double
// buffer, 12x4 WMMA tiles per 64-token sub-chunk, strided scatter to (B,C,N,3).
// ---------------------------------------------------------------------------
__global__ __launch_bounds__(256) void k_apply(const float* __restrict__ feat,
                                               const float* __restrict__ U,
                                               float* __restrict__ out)
{
    __shared__ float Ul[Rn * Rn];         // 147456 B
    __shared__ float Fb[2][Cn][3 * FS];   //  98304 B
    const int b = blockIdx.y, chunk = blockIdx.x, tid = threadIdx.x;
    const int wv = tid >> 5, lane = tid & 31, half = lane >> 4, l16 = lane & 15;
    const int waveR = wv >> 1, waveC = wv & 1;
    const int fchunk = Nn / CH3;                        // 512
    const unsigned long long rowstride = (unsigned long long)Nn * 3;

    {   // load U tile (vectorized, coalesced)
        const float4* src = (const float4*)(U + (size_t)b * Rn * Rn);
        float4* dst = (float4*)Ul;
        for (int idx = tid; idx < (Rn * Rn) / 4; idx += 256) dst[idx] = src[idx];
    }
    if (wv == 0) {
        const float* g = feat + (((size_t)b * Cn) * Nn + (size_t)(chunk * fchunk)) * 3;
        tdm_load_2d((unsigned int)(uintptr_t)&Fb[0][0][0], g, 3 * FS, Cn, rowstride);
    }
    for (int s = 0; s < SUB3; ++s) {
        if (wv == 0) {
            if (s + 1 < SUB3) {
                const float* g = feat +
                    (((size_t)b * Cn) * Nn + (size_t)(chunk * fchunk + (s + 1) * FS)) * 3;
                tdm_load_2d((unsigned int)(uintptr_t)&Fb[(s + 1) & 1][0][0],
                            g, 3 * FS, Cn, rowstride);
                __builtin_amdgcn_s_wait_tensorcnt(1);
            } else {
                __builtin_amdgcn_s_wait_tensorcnt(0);
            }
        }
        __syncthreads();
        const float (*F)[3 * FS] = Fb[s & 1];
        v8f acc[3][2];
        #pragma unroll
        for (int i = 0; i < 3; ++i)
            #pragma unroll
            for (int j = 0; j < 2; ++j)
                acc[i][j] = (v8f){0.f, 0.f, 0.f, 0.f, 0.f, 0.f, 0.f, 0.f};
        #pragma unroll 2
        for (int k0 = 0; k0 < Rn; k0 += 4) {
            const int kk = k0 + 2 * half;
            v2f a[3], bf[2];
            #pragma unroll
            for (int i = 0; i < 3; ++i) {
                const int row = (waveR * 3 + i) * 16 + l16;
                a[i].x = Ul[row * Rn + kk];
                a[i].y = Ul[row * Rn + kk + 1];
            }
            #pragma unroll
            for (int j = 0; j < 2; ++j) {
                const int fl = (waveC * 2 + j) * 16 + l16;
                { const int m = kk >> 6,       ci = kk & 63;       bf[j].x = F[ci][fl * 3 + m]; }
                { const int m = (kk + 1) >> 6, ci = (kk + 1) & 63; bf[j].y = F[ci][fl * 3 + m]; }
            }
            #pragma unroll
            for (int i = 0; i < 3; ++i)
                #pragma unroll
                for (int j = 0; j < 2; ++j)
                    acc[i][j] = __builtin_amdgcn_wmma_f32_16x16x4_f32(
                        false, a[i], false, bf[j], (short)0, acc[i][j], false, false);
        }
        #pragma unroll
        for (int i = 0; i < 3; ++i)
            #pragma unroll
            for (int j = 0; j < 2; ++j) {
                const int fl = (waveC * 2 + j) * 16 + l16;
                const int f  = chunk * fchunk + s * FS + fl;
                #pragma unroll
                for (int v = 0; v < 8; ++v) {
                    const int row = (waveR * 3 + i) * 16 + 8 * half + v;
                    const int n = row >> 6, co = row & 63;
                    out[((size_t)(b * Cn + co) * Nn + f) * 3 + n] = acc[i][j][v];
                }
            }
        __syncthreads();
    }
}

// ---------------------------------------------------------------------------
extern "C" void kernel_launch(void* const* d_in, const int* in_sizes, int n_in,
                              void* d_out, int out_size, void* d_ws, size_t ws_size,
                              hipStream_t stream)
{
    (void)in_sizes; (void)n_in; (void)out_size; (void)ws_size;
    const float* feat = (const float*)d_in[0];
    // d_in[1] = xyz : unused by the reference computation
    const float* Wq   = (const float*)d_in[2];
    const float* Wk   = (const float*)d_in[3];
    const float* Wv   = (const float*)d_in[4];
    const float* Wout = (const float*)d_in[5];
    float* out = (float*)d_out;

    float* Spart = (float*)d_ws;                              // B*CH1*192*192 f32
    float* U     = Spart + (size_t)Bn * CH1 * Rn * Rn;        // B*192*192 f32

    k_gram  <<<dim3(CH1, Bn), 256, 0, stream>>>(feat, Spart);
    k_attn_u<<<Bn,            256, 0, stream>>>(Spart, Wq, Wk, Wv, Wout, U);
    k_apply <<<dim3(CH3, Bn), 256, 0, stream>>>(feat, U, out);
}